// SE3HyperHyenaOperator_34308198761146
// MI455X (gfx1250) — compile-verified
//
#include <hip/hip_runtime.h>
#include <cstdint>

// ---------------------------------------------------------------------------
// SE(3) HyperHyena operator for MI455X (gfx1250, wave32, WMMA, TDM, 320KB LDS).
//
// Pipeline (matches the JAX reference, fp32 throughout):
//   0) init        : CG tensors computed on-device + stats zeroed
//   1) bn_stats    : per-(l,channel) mean ||x_l||^2 over (b,n)   [atomics]
//   2) proj_qkv    : TDM slab->LDS, BN scale, q/k/v 16x16 channel mix via
//                    V_WMMA_F32_16X16X4_F32, per-(l,c) unit normalization
//   3) fft_fwd     : TDM line->LDS, radix-2 LDS FFT (N=8192) of q,k
//   4) couple<L>   : S_m(f) = sum_ij C[i,j,m] Q_i(f) conj(K_j(f))  (in place)
//   5) ifft        : inverse LDS FFT of S -> attn (real part)
//   6) pointwise<L>: out_m = sum_ij C[i,j,m] attn_i v_j per token
//   7) proj_out    : TDM slab->LDS, 16x16 output mix via WMMA -> d_out
//
// Feature layout (flat 256 = concat_l [mult=16, 2l+1], mult-major):
//   chan(l,c,m) = off_l + c*(2l+1) + m,  off_l = {0,16,64,144}
// ---------------------------------------------------------------------------

#define NSEQ   8192
#define NBATCH 8
#define DFEAT  256

typedef float v2f __attribute__((ext_vector_type(2)));
typedef float v8f __attribute__((ext_vector_type(8)));
typedef unsigned int v4u __attribute__((ext_vector_type(4)));
typedef int v4i __attribute__((ext_vector_type(4)));
typedef int v8i __attribute__((ext_vector_type(8)));

__constant__ int d_lm_l[16] = {0,1,1,1,2,2,2,2,2,3,3,3,3,3,3,3};
__constant__ int d_lm_m[16] = {0,0,1,2,0,1,2,3,4,0,1,2,3,4,5,6};

__device__ __forceinline__ int chan_off(int l) {
  return (l == 0) ? 0 : ((l == 1) ? 16 : ((l == 2) ? 64 : 144));
}
__device__ __forceinline__ int cg_base(int l) {
  return (l == 0) ? 0 : ((l == 1) ? 1 : ((l == 2) ? 28 : 153));
}
__device__ __forceinline__ void chan_decode(int ch, int& l, int& c, int& m) {
  if      (ch < 16)  { l = 0; c = ch;          m = 0; }
  else if (ch < 64)  { l = 1; c = (ch-16)/3;   m = (ch-16)%3; }
  else if (ch < 144) { l = 2; c = (ch-64)/5;   m = (ch-64)%5; }
  else               { l = 3; c = (ch-144)/7;  m = (ch-144)%7; }
}

// D = A(16x4) * B(4x16) + C(16x16), fp32, one wave (EXEC must be all ones).
// Fragment layout (ISA 7.12.2, 32-bit):
//   A: lane(h*16+r), vgpr j -> A[row=r][K=j+2h]
//   B: lane(h*16+r), vgpr j -> B[K=j+2h][col=r]
//   C/D: lane(h*16+r), vgpr v -> D[row=v+8h][col=r]
__device__ __forceinline__ v8f wmma_16x16x4_f32(v2f a, v2f b, v8f c) {
  return __builtin_amdgcn_wmma_f32_16x16x4_f32(
      /*neg_a=*/false, a, /*neg_b=*/false, b,
      /*c_mod=*/(short)0, c, /*reuse_a=*/false, /*reuse_b=*/false);
}

// ---------------- Tensor Data Mover: contiguous global -> LDS copy ----------
// D# per ISA 8.3/8.4: group0 = {count=1 | lds_addr | global_addr | type=2},
// group1 = {data_size=4B, tensor_dim0=tile_dim0=nelem, 1 row}. Groups 2/3
// (and the trailing group on this toolchain's 6-arg builtin) are zero for a
// <=2D tensor. Issued by one wave; completion via TENSORcnt.
__device__ __forceinline__ void tdm_load_1d(void* lds_dst, const void* gsrc,
                                            unsigned nelem /* fp32 elements */) {
  const unsigned lds_addr = (unsigned)(uintptr_t)lds_dst;   // low 32 bits = LDS offset
  const unsigned long long ga = (unsigned long long)(uintptr_t)gsrc;
  v4u g0;
  g0[0] = 1u;                                            // count=1 (valid user D#)
  g0[1] = lds_addr;                                      // lds_addr[31:0]
  g0[2] = (unsigned)(ga & 0xffffffffull);                // global_addr[31:0]
  g0[3] = (unsigned)((ga >> 32) & 0x01ffffffull)         // global_addr[56:32]
        | (2u << 30);                                    // type = 2 ("image")
  v8i g1;
  g1[0] = 2 << 16;                                       // data_size = 4 bytes
  g1[1] = (int)((nelem & 0xffffu) << 16);                // tensor_dim0[15:0]<<16
  g1[2] = (int)((nelem >> 16) | (1u << 16));             // tensor_dim0[31:16] | tensor_dim1=1
  g1[3] = (int)((nelem & 0xffffu) << 16);                // tile_dim0 = nelem
  g1[4] = 0;                                             // tile_dim1/2 unused
  g1[5] = (int)nelem;                                    // tensor_dim0_stride[31:0]
  g1[6] = 0;
  g1[7] = 0;
  v4i gz4 = {0, 0, 0, 0};
  v8i gz8 = {0, 0, 0, 0, 0, 0, 0, 0};
  __builtin_amdgcn_tensor_load_to_lds(g0, g1, gz4, gz4, gz8, 0);
}

// ------------------------- CG tensor on device ------------------------------
__device__ double dfact(int n) {
  double r = 1.0;
  for (int i = 2; i <= n; ++i) r *= (double)i;
  return r;
}

// <l m1 l m2 | l (m1+m2)> real CG coefficient (Condon-Shortley), l1=l2=L=l.
__device__ float cg_ll_l(int l, int m1, int m2) {
  const int L = l, M = m1 + m2;
  if (M < -L || M > L) return 0.0f;
  double pre = sqrt((2.0*L + 1.0) * dfact(l+l-L) * dfact(l-l+L) * dfact(-l+l+L)
                    / dfact(l+l+L+1));
  pre *= sqrt(dfact(L+M) * dfact(L-M) * dfact(l-m1) * dfact(l+m1)
              * dfact(l-m2) * dfact(l+m2));
  double s = 0.0;
  for (int k = 0; k <= l+l-L; ++k) {
    const int a0 = k, a1 = l+l-L-k, a2 = l-m1-k, a3 = l+m2-k,
              a4 = L-l+m1+k, a5 = L-l-m2+k;
    if (a0 < 0 || a1 < 0 || a2 < 0 || a3 < 0 || a4 < 0 || a5 < 0) continue;
    const double t = 1.0 / (dfact(a0)*dfact(a1)*dfact(a2)*dfact(a3)*dfact(a4)*dfact(a5));
    s += (k & 1) ? -t : t;
  }
  return (float)(pre * s);
}

__global__ __launch_bounds__(256) void init_kernel(float* __restrict__ cg,
                                                   float* __restrict__ stats) {
  const int t = threadIdx.x;
  if (t < 64) stats[t] = 0.0f;
  for (int idx = t; idx < 496; idx += 256) {
    int l, rel;
    if      (idx < 1)   { l = 0; rel = idx;      }
    else if (idx < 28)  { l = 1; rel = idx - 1;  }
    else if (idx < 153) { l = 2; rel = idx - 28; }
    else                { l = 3; rel = idx - 153;}
    const int d = 2*l + 1;
    const int m = rel % d;
    const int j = (rel / d) % d;
    const int i = rel / (d*d);
    const int m1 = i - l, m2 = j - l, M = m - l;
    cg[idx] = (m1 + m2 == M) ? cg_ll_l(l, m1, m2) : 0.0f;
  }
}

// ------------------------- BN statistics ------------------------------------
__global__ __launch_bounds__(256) void bn_stats_kernel(const float* __restrict__ x,
                                                       float* __restrict__ stats) {
  __shared__ float sbin[64];
  if (threadIdx.x < 64) sbin[threadIdx.x] = 0.0f;
  __syncthreads();
  // stride is a multiple of DFEAT => each thread owns one fixed feature column
  int l, c, m;
  chan_decode(threadIdx.x & (DFEAT - 1), l, c, m);
  float acc = 0.0f;
  const size_t total  = (size_t)NBATCH * NSEQ * DFEAT;
  const size_t stride = (size_t)gridDim.x * 256;
  for (size_t idx = (size_t)blockIdx.x * 256 + threadIdx.x; idx < total; idx += stride) {
    __builtin_prefetch(&x[idx + stride], 0, 0);
    const float vv = x[idx];
    acc += vv * vv;
  }
  atomicAdd(&sbin[l*16 + c], acc);   // ds_add_f32
  __syncthreads();
  if (threadIdx.x < 64) atomicAdd(&stats[threadIdx.x], sbin[threadIdx.x]);
}

// ------------------------- q/k/v projection (WMMA) --------------------------
__global__ __launch_bounds__(256) void proj_qkv_kernel(
    const float* __restrict__ x, const float* __restrict__ bn_w,
    const float* __restrict__ stats,
    const float* __restrict__ qW, const float* __restrict__ kW,
    const float* __restrict__ vW,
    float* __restrict__ qT, float* __restrict__ kT, float* __restrict__ vOut,
    int b) {
  __shared__ float xs[16 * DFEAT];        // BN-scaled input slab, 16 tokens
  __shared__ float ps[3][16 * DFEAT];     // projected q/k/v (pre-normalization)
  const int tokenBase = blockIdx.x * 16;
  const float* xrow = x + ((size_t)(b * NSEQ + tokenBase)) * DFEAT;
  __builtin_prefetch(xrow + 16 * DFEAT, 0, 3);   // next slab -> L2

  // Phase 1: TDM bulk copy slab -> LDS, then BN scale in place
  if (threadIdx.x < 32) {
    tdm_load_1d(xs, xrow, 16 * DFEAT);
    __builtin_amdgcn_s_wait_tensorcnt(0);
  }
  __syncthreads();
  for (int i = threadIdx.x; i < 16 * DFEAT; i += 256) {
    const int ch = i & (DFEAT - 1);
    int l, c, m;
    chan_decode(ch, l, c, m);
    const float n2 = stats[l*16 + c] * (1.0f / ((float)NBATCH * (float)NSEQ));
    const float s  = bn_w[l*16 + c] * __frsqrt_rn(n2 + 1e-6f);
    xs[i] *= s;
  }
  __syncthreads();

  // Phase 2: 48 tasks = {q,k,v} x 16 (l,m) pairs; each task = W(16x16)*X(16x16)
  // readfirstlane keeps the task selection scalar -> EXEC stays all-ones
  // around the WMMAs (required by ISA 7.12).
  const int wv = threadIdx.x >> 5;
  const int lane = threadIdx.x & 31;
  const int r = lane & 15, h = lane >> 4;
  const float* Ws[3] = {qW, kW, vW};
  for (int task = wv; task < 48; task += 8) {
    const int t   = __builtin_amdgcn_readfirstlane(task);
    const int mat = t >> 4;
    const int p   = t & 15;
    const int l   = d_lm_l[p], m = d_lm_m[p];
    const int dd  = 2*l + 1;
    const int base = chan_off(l);
    const float* W = Ws[mat] + l * DFEAT;   // [o=16][i=16]
    v8f acc = {0.f,0.f,0.f,0.f,0.f,0.f,0.f,0.f};
#pragma unroll
    for (int c4 = 0; c4 < 4; ++c4) {        // K = 16 in 4 chunks of 4
      const int i0 = 4*c4 + 2*h;
      v2f a;  a.x  = W[r*16 + i0];                  a.y  = W[r*16 + i0 + 1];
      v2f bb; bb.x = xs[r*DFEAT + base + i0*dd + m];
              bb.y = xs[r*DFEAT + base + (i0+1)*dd + m];
      acc = wmma_16x16x4_f32(a, bb, acc);
    }
#pragma unroll
    for (int vr = 0; vr < 8; ++vr) {
      const int o = vr + 8*h;
      ps[mat][r*DFEAT + base + o*dd + m] = acc[vr];
    }
  }
  __syncthreads();

  // Phase 3: per-(l,c) m-vector normalization + layout-aware stores
  for (int item = threadIdx.x; item < 3 * 16 * 64; item += 256) {
    const int mat   = item / 1024;
    const int rem   = item - mat * 1024;
    const int token = rem >> 6;
    const int pr    = rem & 63;
    const int l = pr >> 4, c = pr & 15;
    const int dd = 2*l + 1;
    const int base = chan_off(l) + c * dd;
    const float* src = &ps[mat][token * DFEAT];
    float pv[7];
    float ss = 0.0f;
    for (int m = 0; m < dd; ++m) { pv[m] = src[base + m]; ss += pv[m]*pv[m]; }
    const float inv = 1.0f / (sqrtf(ss) + 1e-6f);
    const int n = tokenBase + token;
    for (int m = 0; m < dd; ++m) {
      const float val = pv[m] * inv;
      if      (mat == 0) qT[(size_t)(base + m) * NSEQ + n] = val;   // channel-major
      else if (mat == 1) kT[(size_t)(base + m) * NSEQ + n] = val;   // channel-major
      else               vOut[(size_t)n * DFEAT + base + m] = val;  // token-major
    }
  }
}

// ------------------------- forward FFT (q and k) ----------------------------
__global__ __launch_bounds__(256) void fft_fwd_kernel(
    const float* __restrict__ qT, const float* __restrict__ kT,
    float2* __restrict__ Qf, float2* __restrict__ Kf) {
  __shared__ float re[NSEQ];
  __shared__ float im[NSEQ];
  const int bid = blockIdx.x;
  const int ch  = bid & (DFEAT - 1);
  const float* src = ((bid < DFEAT) ? qT : kT) + (size_t)ch * NSEQ;
  float2* dst      = ((bid < DFEAT) ? Qf : Kf) + (size_t)ch * NSEQ;

  // TDM bulk copy of the 32KB sequence line into im[] (staging), then
  // bit-reverse scatter into re[] and rezero im[].
  if (threadIdx.x < 32) {
    tdm_load_1d(im, src, NSEQ);
    __builtin_amdgcn_s_wait_tensorcnt(0);
  }
  __syncthreads();
  for (int j = threadIdx.x; j < NSEQ; j += 256) {
    const int rj = (int)(__brev((unsigned)j) >> 19);   // 13-bit reversal
    re[rj] = im[j];
  }
  __syncthreads();
  for (int j = threadIdx.x; j < NSEQ; j += 256) im[j] = 0.0f;
  __syncthreads();

#pragma unroll 1
  for (int s = 0; s < 13; ++s) {
    const int half = 1 << s;
    const float ang0 = -6.283185307179586f / (float)(half << 1);
    for (int idx = threadIdx.x; idx < NSEQ/2; idx += 256) {
      const int off = idx & (half - 1);
      const int pos = ((idx >> s) << (s + 1)) + off;
      const int oth = pos + half;
      float sa, ca;
      __sincosf(ang0 * (float)off, &sa, &ca);
      const float vr2 = re[oth], vi2 = im[oth];
      const float tr = vr2*ca - vi2*sa;
      const float ti = vr2*sa + vi2*ca;
      const float ur = re[pos], ui = im[pos];
      re[pos] = ur + tr; im[pos] = ui + ti;
      re[oth] = ur - tr; im[oth] = ui - ti;
    }
    __syncthreads();
  }
  for (int j = threadIdx.x; j < NSEQ; j += 256) dst[j] = make_float2(re[j], im[j]);
}

// ------------------------- frequency-domain CG coupling ---------------------
template <int L>
__global__ __launch_bounds__(256) void couple_kernel(
    float2* __restrict__ Qf, const float2* __restrict__ Kf,
    const float* __restrict__ cg) {
  constexpr int D = 2*L + 1;
  __shared__ float sc[D*D*D];
  for (int i = threadIdx.x; i < D*D*D; i += 256) sc[i] = cg[cg_base(L) + i];
  __syncthreads();
  const int g = blockIdx.x * 256 + threadIdx.x;   // 16 channels x 8192 freqs
  const int f = g & (NSEQ - 1);
  const int c = g >> 13;
  const int base = chan_off(L) + c * D;
  float2 Q[D], K[D];
  float Sre[D], Sim[D];
#pragma unroll
  for (int i = 0; i < D; ++i) {
    Q[i] = Qf[(size_t)(base + i) * NSEQ + f];
    K[i] = Kf[(size_t)(base + i) * NSEQ + f];
  }
#pragma unroll
  for (int m = 0; m < D; ++m) { Sre[m] = 0.0f; Sim[m] = 0.0f; }
#pragma unroll
  for (int i = 0; i < D; ++i)
#pragma unroll
    for (int j = 0; j < D; ++j) {
      const float pre = Q[i].x * K[j].x + Q[i].y * K[j].y;   // Q_i * conj(K_j)
      const float pim = Q[i].y * K[j].x - Q[i].x * K[j].y;
#pragma unroll
      for (int m = 0; m < D; ++m) {
        const float cc = sc[(i*D + j)*D + m];
        Sre[m] += cc * pre;
        Sim[m] += cc * pim;
      }
    }
#pragma unroll
  for (int m = 0; m < D; ++m)
    Qf[(size_t)(base + m) * NSEQ + f] = make_float2(Sre[m], Sim[m]);   // in place
}

// ------------------------- inverse FFT -> attn (real) -----------------------
__global__ __launch_bounds__(256) void ifft_kernel(const float2* __restrict__ Sf,
                                                   float* __restrict__ attn) {
  __shared__ float re[NSEQ];
  __shared__ float im[NSEQ];
  const int ch = blockIdx.x;
  const float2* src = Sf + (size_t)ch * NSEQ;
  __builtin_prefetch(src, 0, 3);
  for (int j = threadIdx.x; j < NSEQ; j += 256) {
    const int rj = (int)(__brev((unsigned)j) >> 19);
    const float2 vv = src[j];
    re[rj] = vv.x;
    im[rj] = vv.y;
  }
  __syncthreads();
#pragma unroll 1
  for (int s = 0; s < 13; ++s) {
    const int half = 1 << s;
    const float ang0 = 6.283185307179586f / (float)(half << 1);   // +i for ifft
    for (int idx = threadIdx.x; idx < NSEQ/2; idx += 256) {
      const int off = idx & (half - 1);
      const int pos = ((idx >> s) << (s + 1)) + off;
      const int oth = pos + half;
      float sa, ca;
      __sincosf(ang0 * (float)off, &sa, &ca);
      const float vr2 = re[oth], vi2 = im[oth];
      const float tr = vr2*ca - vi2*sa;
      const float ti = vr2*sa + vi2*ca;
      const float ur = re[pos], ui = im[pos];
      re[pos] = ur + tr; im[pos] = ui + ti;
      re[oth] = ur - tr; im[oth] = ui - ti;
    }
    __syncthreads();
  }
  const float scale = 1.0f / (float)NSEQ;
  for (int j = threadIdx.x; j < NSEQ; j += 256)
    attn[(size_t)j * DFEAT + ch] = re[j] * scale;   // token-major real part
}

// ------------------------- pointwise CG with v ------------------------------
template <int L>
__global__ __launch_bounds__(256) void pointwise_kernel(
    const float* __restrict__ attn, const float* __restrict__ v,
    float* __restrict__ tmp, const float* __restrict__ cg) {
  constexpr int D = 2*L + 1;
  __shared__ float sc[D*D*D];
  for (int i = threadIdx.x; i < D*D*D; i += 256) sc[i] = cg[cg_base(L) + i];
  __syncthreads();
  const int g = blockIdx.x * 256 + threadIdx.x;   // 8192 tokens x 16 channels
  const int c = g & 15;
  const int n = g >> 4;
  const int base = chan_off(L) + c * D;
  float A[D], V[D], O[D];
#pragma unroll
  for (int i = 0; i < D; ++i) {
    A[i] = attn[(size_t)n * DFEAT + base + i];
    V[i] = v[(size_t)n * DFEAT + base + i];
    O[i] = 0.0f;
  }
#pragma unroll
  for (int i = 0; i < D; ++i)
#pragma unroll
    for (int j = 0; j < D; ++j) {
      const float p = A[i] * V[j];
#pragma unroll
      for (int m = 0; m < D; ++m) O[m] += sc[(i*D + j)*D + m] * p;
    }
#pragma unroll
  for (int m = 0; m < D; ++m) tmp[(size_t)n * DFEAT + base + m] = O[m];
}

// ------------------------- output projection (WMMA) -------------------------
__global__ __launch_bounds__(256) void proj_out_kernel(
    const float* __restrict__ tmp, const float* __restrict__ oW,
    float* __restrict__ out, int b) {
  __shared__ float xs[16 * DFEAT];
  const int tokenBase = blockIdx.x * 16;
  const float* trow = tmp + (size_t)tokenBase * DFEAT;
  __builtin_prefetch(trow + 16 * DFEAT, 0, 3);

  // TDM bulk copy slab -> LDS (pure linear copy, no post-processing)
  if (threadIdx.x < 32) {
    tdm_load_1d(xs, trow, 16 * DFEAT);
    __builtin_amdgcn_s_wait_tensorcnt(0);
  }
  __syncthreads();

  const int wv = threadIdx.x >> 5, lane = threadIdx.x & 31;
  const int r = lane & 15, h = lane >> 4;
  float* orow = out + ((size_t)(b * NSEQ + tokenBase)) * DFEAT;
  for (int p = wv; p < 16; p += 8) {
    const int pU = __builtin_amdgcn_readfirstlane(p);
    const int l = d_lm_l[pU], m = d_lm_m[pU];
    const int dd = 2*l + 1;
    const int base = chan_off(l);
    const float* W = oW + l * DFEAT;
    v8f acc = {0.f,0.f,0.f,0.f,0.f,0.f,0.f,0.f};
#pragma unroll
    for (int c4 = 0; c4 < 4; ++c4) {
      const int i0 = 4*c4 + 2*h;
      v2f a;  a.x  = W[r*16 + i0];                    a.y  = W[r*16 + i0 + 1];
      v2f bb; bb.x = xs[r*DFEAT + base + i0*dd + m];
              bb.y = xs[r*DFEAT + base + (i0+1)*dd + m];
      acc = wmma_16x16x4_f32(a, bb, acc);
    }
#pragma unroll
    for (int vr = 0; vr < 8; ++vr) {
      const int o = vr + 8*h;
      orow[r*DFEAT + base + o*dd + m] = acc[vr];
    }
  }
}

// ------------------------- host launcher ------------------------------------
extern "C" void kernel_launch(void* const* d_in, const int* in_sizes, int n_in,
                              void* d_out, int out_size, void* d_ws, size_t ws_size,
                              hipStream_t stream) {
  (void)in_sizes; (void)n_in; (void)out_size; (void)ws_size;
  const float* x    = (const float*)d_in[0];
  const float* bn_w = (const float*)d_in[1];
  const float* qW   = (const float*)d_in[2];
  const float* kW   = (const float*)d_in[3];
  const float* vW   = (const float*)d_in[4];
  const float* oW   = (const float*)d_in[5];
  float* out = (float*)d_out;

  // Workspace layout (per-batch buffers reused across the b loop; ~56 MB)
  char* ws = (char*)d_ws;
  float* cg_d  = (float*)(ws + 0);      // 496 floats (CG tensors, all l)
  float* stats = (float*)(ws + 2048);   // 64 floats  (BN sum of squares)
  size_t off = 4096;
  float*  qT = (float*)(ws + off); off += (size_t)DFEAT * NSEQ * sizeof(float);   // 8 MB
  float*  kT = (float*)(ws + off); off += (size_t)DFEAT * NSEQ * sizeof(float);   // 8 MB
  float*  vB = (float*)(ws + off); off += (size_t)DFEAT * NSEQ * sizeof(float);   // 8 MB
  float2* Qf = (float2*)(ws + off); off += (size_t)DFEAT * NSEQ * sizeof(float2); // 16 MB
  float2* Kf = (float2*)(ws + off); off += (size_t)DFEAT * NSEQ * sizeof(float2); // 16 MB

  init_kernel<<<1, 256, 0, stream>>>(cg_d, stats);
  bn_stats_kernel<<<1024, 256, 0, stream>>>(x, stats);

  for (int b = 0; b < NBATCH; ++b) {
    proj_qkv_kernel<<<NSEQ/16, 256, 0, stream>>>(x, bn_w, stats, qW, kW, vW,
                                                 qT, kT, vB, b);
    fft_fwd_kernel<<<2*DFEAT, 256, 0, stream>>>(qT, kT, Qf, Kf);
    couple_kernel<0><<<16*NSEQ/256, 256, 0, stream>>>(Qf, Kf, cg_d);
    couple_kernel<1><<<16*NSEQ/256, 256, 0, stream>>>(Qf, Kf, cg_d);
    couple_kernel<2><<<16*NSEQ/256, 256, 0, stream>>>(Qf, Kf, cg_d);
    couple_kernel<3><<<16*NSEQ/256, 256, 0, stream>>>(Qf, Kf, cg_d);
    ifft_kernel<<<DFEAT, 256, 0, stream>>>(Qf, kT);              // attn -> kT (token-major)
    pointwise_kernel<0><<<16*NSEQ/256, 256, 0, stream>>>(kT, vB, qT, cg_d);
    pointwise_kernel<1><<<16*NSEQ/256, 256, 0, stream>>>(kT, vB, qT, cg_d);
    pointwise_kernel<2><<<16*NSEQ/256, 256, 0, stream>>>(kT, vB, qT, cg_d);
    pointwise_kernel<3><<<16*NSEQ/256, 256, 0, stream>>>(kT, vB, qT, cg_d);
    proj_out_kernel<<<NSEQ/16, 256, 0, stream>>>(qT, oW, out, b);
  }
}